// WDMChannelMapper_67001489817665
// MI455X (gfx1250) — compile-verified
//
#include <hip/hip_runtime.h>
#include <hip/hip_bf16.h>
#include <math.h>

typedef __attribute__((ext_vector_type(2))) float v2f;
typedef __attribute__((ext_vector_type(4))) float v4f;
typedef __attribute__((ext_vector_type(8))) float v8f;

// ---------------------------------------------------------------------------
// Kernel 1: one wave32. Computes
//   r_c   = g_c * sigmoid(aw_c) * cos(phase_c + 0.5*D*(wl_c - mean(wl))^2)
//   t_k   = sum_c r_c * M[c,k]          <-- the reference's 'bcf,ck->bkf'
//   alpha = sum_k softmax(aw)_k * t_k
// The t = r^T * M contraction is done with V_WMMA_F32_16X16X4_F32:
//   A (16x4 f32, documented layout): row 0 = [r0 r1 r2 0], rest 0
//   B (4x16 f32): B[c,n] = M[c*3+n] for c<3,n<3, else 0
//   D (16x16 f32): D[0,n] = t_n  -> VGPR0, lanes 0..2
// ---------------------------------------------------------------------------
__global__ void wdm_alpha_wmma_kernel(const float* __restrict__ gains,
                                      const float* __restrict__ phases,
                                      const float* __restrict__ aw,
                                      const float* __restrict__ dcoef,
                                      const float* __restrict__ M,
                                      const float* __restrict__ wl,
                                      float* __restrict__ alpha_out) {
  const int lane = (int)(threadIdx.x & 31u);
  const int h    = lane >> 4;   // lane half
  const int lm   = lane & 15;   // lane within half

  const float wl0 = wl[0], wl1 = wl[1], wl2 = wl[2];
  const float wmean = (wl0 + wl1 + wl2) * (1.0f / 3.0f);
  const float D = dcoef[0];

  const float aw0 = aw[0], aw1 = aw[1], aw2 = aw[2];

  // r_c = gain_c * sigmoid(aw_c) * cos(phase_c + disp_c)
  float d0 = wl0 - wmean, d1 = wl1 - wmean, d2 = wl2 - wmean;
  float p0 = phases[0] + 0.5f * D * d0 * d0;
  float p1 = phases[1] + 0.5f * D * d1 * d1;
  float p2 = phases[2] + 0.5f * D * d2 * d2;
  float r0 = gains[0] * (1.0f / (1.0f + expf(-aw0))) * cosf(p0);
  float r1 = gains[1] * (1.0f / (1.0f + expf(-aw1))) * cosf(p1);
  float r2 = gains[2] * (1.0f / (1.0f + expf(-aw2))) * cosf(p2);

  // softmax(aw) (max-subtracted, matching jax.nn.softmax)
  float mx = fmaxf(aw0, fmaxf(aw1, aw2));
  float e0 = expf(aw0 - mx), e1 = expf(aw1 - mx), e2 = expf(aw2 - mx);
  float inv = 1.0f / (e0 + e1 + e2);
  float s0 = e0 * inv, s1 = e1 * inv, s2 = e2 * inv;

  // A matrix (16x4 f32): lane L (half h, m=L&15): vgpr j holds A[m, 2h+j].
  // Row 0 = r, all other rows 0. Branch-free (EXEC must stay all-ones).
  v2f a;
  a.x = (lm == 0) ? ((h == 0) ? r0 : r2) : 0.0f;   // K = 2h
  a.y = (lm == 0) ? ((h == 0) ? r1 : 0.0f) : 0.0f; // K = 2h+1 (K=3 -> 0)

  // B matrix (4x16 f32): vgpr j, lane-half h hold row k = 2h+j, column lm.
  // B[c,n] = M[c*3+n] for c<3 && n<3.
  float m_b0 = (lm < 3) ? M[(2 * h) * 3 + lm] : 0.0f;         // rows 0 / 2
  float m_b1 = (lm < 3 && h == 0) ? M[1 * 3 + lm] : 0.0f;     // row 1 (row 3 = 0)
  v2f b;
  b.x = m_b0;
  b.y = m_b1;

  v8f c = {};
  // D = A x B + 0  (emits v_wmma_f32_16x16x4_f32)
  v8f d = __builtin_amdgcn_wmma_f32_16x16x4_f32(
      /*neg_a=*/false, a, /*neg_b=*/false, b,
      /*c_mod=*/(short)0, c, /*reuse_a=*/false, /*reuse_b=*/false);

  // t_n = D[0,n] lives in D VGPR0 at lane n (n < 16).
  float dv = d[0];
  float t0v = __shfl(dv, 0, 32);
  float t1v = __shfl(dv, 1, 32);
  float t2v = __shfl(dv, 2, 32);
  float alpha = s0 * t0v + s1 * t1v + s2 * t2v;

  if (lane == 0) alpha_out[0] = alpha;
}

// ---------------------------------------------------------------------------
// Kernel 2: bulk path. Pure bandwidth problem: out = alpha * x over 16.7M f32.
// b128 non-temporal loads/stores, grid-stride.
// ---------------------------------------------------------------------------
__global__ void wdm_scale_kernel(const v4f* __restrict__ x,
                                 v4f* __restrict__ out,
                                 const float* __restrict__ alpha_p,
                                 int n4) {
  const float alpha = alpha_p[0];
  const int stride = (int)(gridDim.x * blockDim.x);
  for (int i = (int)(blockIdx.x * blockDim.x + threadIdx.x); i < n4; i += stride) {
    v4f v = __builtin_nontemporal_load(&x[i]);
    v4f o;
    o.x = alpha * v.x;
    o.y = alpha * v.y;
    o.z = alpha * v.z;
    o.w = alpha * v.w;
    __builtin_nontemporal_store(o, &out[i]);
  }
}

// Scalar tail (n not divisible by 4 — defensive; N=4096*4096 is exact).
__global__ void wdm_scale_tail_kernel(const float* __restrict__ x,
                                      float* __restrict__ out,
                                      const float* __restrict__ alpha_p,
                                      int start, int n) {
  const float alpha = alpha_p[0];
  int i = start + (int)(blockIdx.x * blockDim.x + threadIdx.x);
  if (i < n) out[i] = alpha * x[i];
}

extern "C" void kernel_launch(void* const* d_in, const int* in_sizes, int n_in,
                              void* d_out, int out_size, void* d_ws, size_t ws_size,
                              hipStream_t stream) {
  const float* x      = (const float*)d_in[0];
  const float* gains  = (const float*)d_in[1];
  const float* phases = (const float*)d_in[2];
  const float* aw     = (const float*)d_in[3];
  const float* dco    = (const float*)d_in[4];
  const float* M      = (const float*)d_in[5];
  const float* wl     = (const float*)d_in[6];
  float* out   = (float*)d_out;
  float* alpha = (float*)d_ws;

  // Stage 1: alpha via WMMA crosstalk contraction (one wave).
  wdm_alpha_wmma_kernel<<<1, 32, 0, stream>>>(gains, phases, aw, dco, M, wl, alpha);

  // Stage 2: stream out = alpha * x.
  const int n  = in_sizes[0];
  const int n4 = n >> 2;
  const int threads = 256;                 // 8 wave32 per block
  const int blocks  = 4096;                // 1M threads -> 4 b128 iters each
  wdm_scale_kernel<<<blocks, threads, 0, stream>>>((const v4f*)x, (v4f*)out, alpha, n4);

  const int tail = n - (n4 << 2);
  if (tail > 0) {
    wdm_scale_tail_kernel<<<1, 64, 0, stream>>>(x, out, alpha, n4 << 2, n);
  }
}